// SparseCompressConvolution_76459007803906
// MI455X (gfx1250) — compile-verified
//
#include <hip/hip_runtime.h>
#include <hip/hip_bf16.h>

typedef __attribute__((ext_vector_type(2))) float v2f;
typedef __attribute__((ext_vector_type(8))) float v8f;

#define N_SEQ   4096
#define D_IN    128
#define E_DIM   192
#define D_MODEL 128
#define K_TOT   576      // 3*128 + 192
#define FEAT_K  384      // 3*128
#define K_STEPS 144      // K_TOT / 4
#define WAVES_PER_BLOCK 16
#define LDS_W_BYTES (K_TOT * D_MODEL * 4)   // 294912 B = 288 KB (fits 320 KB WGP LDS)

// ---------------------------------------------------------------------------
// adj pass-through: pure 64 MB stream copy, float4 vectorized
// ---------------------------------------------------------------------------
__global__ void scc_copy4_kernel(const float4* __restrict__ src,
                                 float4* __restrict__ dst, int n4) {
    int idx = blockIdx.x * blockDim.x + threadIdx.x;
    int stride = gridDim.x * blockDim.x;
    for (; idx < n4; idx += stride) dst[idx] = src[idx];
}

// ---------------------------------------------------------------------------
// m[p,i] = max(mask[p,i..i+2]) with zero pad past N
// ---------------------------------------------------------------------------
__global__ void scc_maskwin_kernel(const float* __restrict__ mask,
                                   float* __restrict__ m_out) {
    int idx = blockIdx.x * blockDim.x + threadIdx.x;   // 0..P*N-1
    int p = idx >> 12;
    int i = idx & (N_SEQ - 1);
    const float* mp = mask + (size_t)p * N_SEQ;
    float v = mp[i];
    if (i + 1 < N_SEQ) v = fmaxf(v, mp[i + 1]);
    if (i + 2 < N_SEQ) v = fmaxf(v, mp[i + 2]);
    m_out[idx] = v;
}

// ---------------------------------------------------------------------------
// Main fused kernel: feat_cat @ W (fp32 WMMA 16x16x4) + bias -> ReLU,
// + jump (3-tap mean), masked writes of res_m and out_m.
// One wave32 = one 16-row x 128-col output tile. W staged in LDS,
// pre-swizzled into B-fragment lane layout so inner loop = ds_load_2addr_b64
// + v_wmma_f32_16x16x4_f32. A-gather is strength-reduced: the (c=k/3, t=k%3)
// decode has period 3 in the K-step, so we precompute 6 per-lane base
// pointers and the inner loop is pure increment addressing, branch-free.
// ---------------------------------------------------------------------------
__global__ __launch_bounds__(512)
void scc_gemm_kernel(const float* __restrict__ features,
                     const float* __restrict__ edgef,
                     const float* __restrict__ mask,
                     const float* __restrict__ W,
                     const float* __restrict__ bias,
                     float* __restrict__ res_m,
                     float* __restrict__ out_m) {
    extern __shared__ float lds_w[];   // K_TOT*D_MODEL floats, pair-swizzled

    const int tid = threadIdx.x;

    // ---- stage W into LDS, swizzled for B fragments --------------------
    // pair q in [0, 36864): s = q>>8, ct = (q>>5)&7, ln = q&31
    //   koff = (ln<16)?0:2 ; col = ct*16 + (ln&15) ; k = 4*s + koff
    //   lds[2q] = W[k][col] ; lds[2q+1] = W[k+1][col]
    for (int j = 0; j < 72; ++j) {
        int q   = j * 512 + tid;
        int s   = q >> 8;
        int ct  = (q >> 5) & 7;
        int ln  = q & 31;
        int koff = (ln < 16) ? 0 : 2;
        int col  = ct * 16 + (ln & 15);
        int k    = 4 * s + koff;
        lds_w[2 * q]     = W[(size_t)k * D_MODEL + col];
        lds_w[2 * q + 1] = W[(size_t)(k + 1) * D_MODEL + col];
    }
    __syncthreads();

    const int wave = tid >> 5;
    const int lane = tid & 31;
    const int tile = blockIdx.x * WAVES_PER_BLOCK + wave;   // 0..4095
    const int b    = tile >> 8;                             // 256 tiles / batch
    const int i0   = (tile & 255) << 4;                     // first row in tile
    const int mrow = lane & 15;                             // A-matrix M index
    const int i    = i0 + mrow;
    const int koff = (lane >= 16) ? 2 : 0;

    // clamped tap rows + 0/1 validity (only last tile of each batch clamps)
    const int row1 = (i + 1 < N_SEQ) ? (i + 1) : (N_SEQ - 1);
    const int row2 = (i + 2 < N_SEQ) ? (i + 2) : (N_SEQ - 1);
    const float* fb0 = features + ((size_t)b * N_SEQ + i) * D_IN;
    const float* fb1 = features + ((size_t)b * N_SEQ + row1) * D_IN;
    const float* fb2 = features + ((size_t)b * N_SEQ + row2) * D_IN;
    const float* fbt[3] = { fb0, fb1, fb2 };
    const float  fvt[3] = { 1.0f,
                            (i + 1 < N_SEQ) ? 1.0f : 0.0f,
                            (i + 2 < N_SEQ) ? 1.0f : 0.0f };

    // per-phase A-gather base pointers: ka = 12j + 4p + koff
    const float* pa0[3]; const float* pa1[3];
    float va0[3], va1[3];
#pragma unroll
    for (int p = 0; p < 3; ++p) {
        int ka = 4 * p + koff;
        int c0 = ka / 3, t0 = ka - 3 * c0;
        int kb = ka + 1;
        int c1 = kb / 3, t1 = kb - 3 * c1;
        pa0[p] = fbt[t0] + c0;  va0[p] = fvt[t0];
        pa1[p] = fbt[t1] + c1;  va1[p] = fvt[t1];
    }

    v8f acc[8];
#pragma unroll
    for (int ct = 0; ct < 8; ++ct) { v8f z = {}; acc[ct] = z; }

    // ---- feature phase: 96 K-steps as 32 triples, branch/div free ------
    for (int j = 0; j < 32; ++j) {
#pragma unroll
        for (int p = 0; p < 3; ++p) {
            const int s = 3 * j + p;
            v2f afrag;
            afrag.x = pa0[p][4 * j] * va0[p];
            afrag.y = pa1[p][4 * j] * va1[p];
            const float* wrow = lds_w + (size_t)s * 512 + lane * 2;
#pragma unroll
            for (int ct = 0; ct < 8; ++ct) {
                v2f bfrag = *(const v2f*)(wrow + ct * 64);
                acc[ct] = __builtin_amdgcn_wmma_f32_16x16x4_f32(
                    false, afrag, false, bfrag, (short)0, acc[ct], false, false);
            }
        }
    }

    // ---- edge phase: 48 K-steps, A fragment contiguous -> 8B load ------
    const float* eb = edgef + ((size_t)b * N_SEQ + i) * E_DIM + koff;
    for (int s = 96; s < K_STEPS; ++s) {
        v2f afrag = *(const v2f*)(eb + 4 * s - FEAT_K);     // global_load_b64
        const float* wrow = lds_w + (size_t)s * 512 + lane * 2;
#pragma unroll
        for (int ct = 0; ct < 8; ++ct) {
            v2f bfrag = *(const v2f*)(wrow + ct * 64);
            acc[ct] = __builtin_amdgcn_wmma_f32_16x16x4_f32(
                false, afrag, false, bfrag, (short)0, acc[ct], false, false);
        }
    }

    // ---- epilogue: bias + ReLU + jump + mask, write res_m / out_m ------
    // D layout: VGPR r, lanes 0-15 -> M=r, lanes 16-31 -> M=r+8; N = lane&15
    const int nc    = lane & 15;
    const int rhalf = (lane >= 16) ? 8 : 0;
    const int rbase = i0 + rhalf;
    const float* maskp = mask + (size_t)(b & 3) * N_SEQ;

    float mvv[8];
#pragma unroll
    for (int r = 0; r < 8; ++r) {
        int irow = rbase + r;
        float mv = maskp[irow];
        if (irow + 1 < N_SEQ) mv = fmaxf(mv, maskp[irow + 1]);
        if (irow + 2 < N_SEQ) mv = fmaxf(mv, maskp[irow + 2]);
        mvv[r] = mv;
    }

    const float* fcol_base = features + (size_t)b * N_SEQ * D_IN;
#pragma unroll
    for (int ct = 0; ct < 8; ++ct) {
        const int col  = ct * 16 + nc;
        const float bc = bias[col];

        // sliding window of 10 feature rows covers jump for all 8 r's
        float fv[10];
#pragma unroll
        for (int u = 0; u < 10; ++u) {
            int rr = rbase + u;
            float val = (rr < N_SEQ) ? 1.0f : 0.0f;
            int rc = (rr < N_SEQ) ? rr : (N_SEQ - 1);
            fv[u] = fcol_base[(size_t)rc * D_IN + col] * val;
        }

#pragma unroll
        for (int r = 0; r < 8; ++r) {
            float o  = fmaxf(acc[ct][r] + bc, 0.0f);
            float jm = (fv[r] + fv[r + 1] + fv[r + 2]) * (1.0f / 3.0f);
            size_t rowidx = ((size_t)b * N_SEQ + rbase + r) * D_MODEL + col;
            out_m[rowidx] = mvv[r] * o;
            res_m[rowidx] = mvv[r] * (o + jm);
        }
    }
}

// ---------------------------------------------------------------------------
extern "C" void kernel_launch(void* const* d_in, const int* in_sizes, int n_in,
                              void* d_out, int out_size, void* d_ws, size_t ws_size,
                              hipStream_t stream) {
    const float* adj   = (const float*)d_in[0];   // (4096,4096)
    const float* feats = (const float*)d_in[1];   // (16,4096,128)
    const float* edgef = (const float*)d_in[2];   // (16,4096,192)
    const float* mask  = (const float*)d_in[3];   // (4,4096,1)
    const float* W     = (const float*)d_in[4];   // (576,128)
    const float* bias  = (const float*)d_in[5];   // (128,)

    float* out      = (float*)d_out;
    float* out_adj  = out;                                   // 16777216
    float* res_m    = out_adj + (size_t)N_SEQ * N_SEQ;       //  8388608
    float* out_m    = res_m + (size_t)16 * N_SEQ * D_MODEL;  //  8388608
    float* m_out    = out_m + (size_t)16 * N_SEQ * D_MODEL;  //    16384

    // adj pass-through: 4194304 float4
    scc_copy4_kernel<<<4096, 256, 0, stream>>>(
        (const float4*)adj, (float4*)out_adj, (int)((size_t)N_SEQ * N_SEQ / 4));

    // m = window-max of mask
    scc_maskwin_kernel<<<64, 256, 0, stream>>>(mask, m_out);

    // fused GEMM + epilogue: 4096 tiles = 256 blocks x 16 waves
    scc_gemm_kernel<<<256, 512, LDS_W_BYTES, stream>>>(
        feats, edgef, mask, W, bias, res_m, out_m);
}